// TopkGELU_15221364097193
// MI455X (gfx1250) — compile-verified
//
#include <hip/hip_runtime.h>

// Row length of the reduction (last dim of x), threads per block, chunks of
// float4 per thread: 4096 = 4 chunks * 256 threads * 4 floats.
#define ROWLEN 4096
#define BLOCK  256
#define CHUNKS 4

typedef float f32x4 __attribute__((ext_vector_type(4)));
typedef __attribute__((address_space(3))) float lds_f32;

// Order-preserving map fp32 -> u32 (larger key <=> larger float).
__device__ __forceinline__ unsigned key_of(float f) {
  unsigned u = __float_as_uint(f);
  return (u & 0x80000000u) ? ~u : (u | 0x80000000u);
}

// Branchless exact-form GELU: 0.5*x*(1+erf(x/sqrt(2))) with erf via the
// Abramowitz-Stegun 7.1.26 rational (|err| <= 1.5e-7, no exec divergence).
// g = 0.5*(x + |x|*erf(|x|/sqrt(2))).
__device__ __forceinline__ float gelu_exact(float x) {
  const float ax = fabsf(x);
  const float z  = ax * 0.7071067811865475f;
  const float t  = __builtin_amdgcn_rcpf(fmaf(0.3275911f, z, 1.0f));
  float p = fmaf(1.061405429f, t, -1.453152027f);
  p = fmaf(p, t, 1.421413741f);
  p = fmaf(p, t, -0.284496736f);
  p = fmaf(p, t, 0.254829592f);
  p = p * t;
  const float e       = __expf(-z * z);          // v_exp_f32 path
  const float erf_abs = fmaf(-p, e, 1.0f);       // erf(z), z >= 0
  return 0.5f * fmaf(ax, erf_abs, x);
}

__global__ __launch_bounds__(BLOCK) void topk_gelu_rowselect(
    const float* __restrict__ x,
    const int*   __restrict__ kptr,
    float*       __restrict__ out) {
  __shared__ float    grow[ROWLEN];   // 16 KB: gelu(x) for this row
  __shared__ unsigned hist[256];      // radix-select histogram
  __shared__ unsigned wtot[8];        // per-wave bin-group totals
  __shared__ unsigned ctrl[2];        // {chosen digit, new krem}

  const int       tid  = threadIdx.x;
  const int       lane = tid & 31;
  const int       wav  = tid >> 5;
  const long long row  = (long long)blockIdx.x;
  const float*    xrow = x   + row * (long long)ROWLEN;
  float*          orow = out + row * (long long)ROWLEN;

  // LDS byte offset of grow[] within this workgroup's LDS allocation.
  const unsigned lbase = (unsigned)(size_t)((lds_f32*)&grow[0]);

  // ---- Stage the row into LDS with the CDNA5 async data mover ------------
  // GVS mode: saddr = uniform row base (SGPR pair), vaddr = per-lane byte
  // offset, vdst = per-lane LDS byte address. Read-once stream -> NT hint.
#pragma unroll
  for (int c = 0; c < CHUNKS; ++c) {
    unsigned boff  = (unsigned)((c * BLOCK + tid) * 16); // bytes into the row
    unsigned laddr = lbase + boff;
    asm volatile("global_load_async_to_lds_b128 %0, %1, %2 th:TH_LOAD_NT"
                 :
                 : "v"(laddr), "v"(boff), "s"(xrow)
                 : "memory");
  }
  asm volatile("s_wait_asynccnt 0x0" ::: "memory");
  __syncthreads();

  // ---- Branchless exact GELU on this thread's 16 elements ----------------
  float g[CHUNKS * 4];
#pragma unroll
  for (int c = 0; c < CHUNKS; ++c) {
    const int e0 = c * (BLOCK * 4) + tid * 4;
    f32x4 v = *(const f32x4*)&grow[e0];
    f32x4 o;
    o.x = gelu_exact(v.x);
    o.y = gelu_exact(v.y);
    o.z = gelu_exact(v.z);
    o.w = gelu_exact(v.w);
    g[c * 4 + 0] = o.x;
    g[c * 4 + 1] = o.y;
    g[c * 4 + 2] = o.z;
    g[c * 4 + 3] = o.w;
    *(f32x4*)&grow[e0] = o;  // own slots only: no cross-thread hazard here
  }
  __syncthreads();

  // ---- Radix select: exact key of the k-th largest g in this row ---------
  const unsigned k = (unsigned)*kptr;
  unsigned prefix = 0, pmask = 0, krem = k;
  for (int r = 0; r < 4; ++r) {
    const int shift = 24 - 8 * r;
    hist[tid] = 0;  // BLOCK == 256 bins
    __syncthreads();
#pragma unroll
    for (int i = 0; i < CHUNKS * 4; ++i) {
      unsigned key = key_of(g[i]);
      if ((key & pmask) == prefix)
        atomicAdd(&hist[(key >> shift) & 0xFFu], 1u);
    }
    __syncthreads();

    // Parallel bin selection: thread t owns bin t. Need the largest bin b
    // with suffix_incl(b) >= krem; then new krem = krem - cnt_greater(b).
    const unsigned cnt = hist[tid];
    unsigned s = cnt;  // intra-wave inclusive suffix sum (5 shuffle steps)
#pragma unroll
    for (int off = 1; off < 32; off <<= 1) {
      unsigned o = __shfl_down(s, off, 32);
      if (lane + off < 32) s += o;
    }
    if (lane == 0) wtot[wav] = s;  // total of this wave's 32 bins
    __syncthreads();
    unsigned hi = 0;
#pragma unroll
    for (int w = 0; w < 8; ++w)
      if (w > wav) hi += wtot[w];
    const unsigned suffix_incl = s + hi;          // count of keys in bins >= tid
    const unsigned cnt_greater = suffix_incl - cnt;  // bins strictly > tid
    if (suffix_incl >= krem && cnt_greater < krem) {  // exactly one thread
      ctrl[0] = (unsigned)tid;
      ctrl[1] = krem - cnt_greater;
    }
    __syncthreads();
    prefix |= ctrl[0] << shift;
    pmask  |= 0xFFu << shift;
    krem    = ctrl[1];
    // ctrl/wtot are not rewritten until after barriers next round: safe.
  }
  const unsigned T = prefix;  // exact key of the k-th largest value

  // ---- Epilogue: keep top-k (lower-index ties win), NT stream out --------
#pragma unroll
  for (int c = 0; c < CHUNKS; ++c) {
    const int e0 = c * (BLOCK * 4) + tid * 4;
    float res[4];
#pragma unroll
    for (int j = 0; j < 4; ++j) {
      const float    gv  = g[c * 4 + j];
      const unsigned key = key_of(gv);
      float r = 0.0f;
      if (key > T) {
        r = gv;
      } else if (key == T) {
        // Deterministic tie-break: keep the krem lowest-indexed ties.
        const int e = e0 + j;
        unsigned rank = 0;
        for (int q = 0; q < e; ++q) rank += (key_of(grow[q]) == T) ? 1u : 0u;
        if (rank < krem) r = gv;
      }
      res[j] = r;
    }
    f32x4 o;
    o.x = res[0]; o.y = res[1]; o.z = res[2]; o.w = res[3];
    __builtin_nontemporal_store(o, (f32x4*)&orow[e0]);  // write-once stream
  }
}

extern "C" void kernel_launch(void* const* d_in, const int* in_sizes, int n_in,
                              void* d_out, int out_size, void* d_ws, size_t ws_size,
                              hipStream_t stream) {
  const float* x    = (const float*)d_in[0];
  const int*   kptr = (const int*)d_in[1];
  float*       out  = (float*)d_out;
  const int rows = in_sizes[0] / ROWLEN;  // (4*4096*4096)/4096 = 16384
  topk_gelu_rowselect<<<dim3((unsigned)rows), dim3(BLOCK), 0, stream>>>(x, kptr, out);
}